// GCN_37306085933361
// MI455X (gfx1250) — compile-verified
//
#include <hip/hip_runtime.h>
#include <hip/hip_bf16.h>

#define NNODES 20000
#define NEDGES 320000
#define DIN    128
#define HDIM   512
#define NOUT   7
#define LNEPS  1e-5f
#define KCHUNK 32            // K-tile staged in LDS per block
#define APAD   34            // 32 + 2 pad: conflict-free ds_load_b64, 8B aligned
#define MPROWS 125           // 160 blocks * 125 rows = 20000

typedef __attribute__((ext_vector_type(2))) float v2f;
typedef __attribute__((ext_vector_type(8))) float v8f;

// ---------------------------------------------------------------------------
// Degree / dinv
// ---------------------------------------------------------------------------
__global__ __launch_bounds__(256) void k_init_deg(float* __restrict__ deg, int n) {
    int i = blockIdx.x * blockDim.x + threadIdx.x;
    if (i < n) deg[i] = 1.0f;            // self loop contributes 1
}

__global__ __launch_bounds__(256) void k_deg_edges(const int* __restrict__ ei,
                                                   float* __restrict__ deg, int e) {
    int t = blockIdx.x * blockDim.x + threadIdx.x;
    if (t < e) unsafeAtomicAdd(&deg[ei[e + t]], 1.0f);   // dst row of edge_index
}

__global__ __launch_bounds__(256) void k_dinv(float* __restrict__ deg, int n) {
    int i = blockIdx.x * blockDim.x + threadIdx.x;
    if (i < n) deg[i] = rsqrtf(deg[i]);  // deg >= 1 always (self loop)
}

// ---------------------------------------------------------------------------
// Dense GEMM via V_WMMA_F32_16X16X4_F32 (exact fp32, D = A*B + C)
// A: [M,K] row-major, B: [K,Ncols] row-major, C: [M,Ncols]
// grid.x = M/16 row tiles; (grid.y * 8 waves) * 16 = Ncols column tiles.
// A chunk [16 x KCHUNK] is staged in LDS once per block (coalesced, shared by
// all 8 waves); fragments then come from LDS via conflict-free ds_load_b64.
// ISA layouts (wave32):
//   A 16x4 : lanes 0-15 hold rows M=0..15; v0={K=k0 | K=k0+2}, v1={k0+1 | k0+3}
//   B 4x16 : lanes 0-15 hold cols N=0..15; v0={K=k0 | K=k0+2}, v1={k0+1 | k0+3}
//   C 16x16: vgpr j, lanes 0-15 -> M=j, lanes 16-31 -> M=j+8, N=lane&15
// ---------------------------------------------------------------------------
__global__ __launch_bounds__(256) void k_gemm_wmma_f32(const float* __restrict__ A,
                                                       const float* __restrict__ B,
                                                       float* __restrict__ C,
                                                       int K, int Ncols) {
    __shared__ float As[16][APAD];

    const int lane = threadIdx.x & 31;
    const int wave = threadIdx.x >> 5;
    const int half = lane >> 4;        // 0: lanes 0-15, 1: lanes 16-31
    const int l16  = lane & 15;

    const int row0 = blockIdx.x * 16;
    const int col0 = (blockIdx.y * 8 + wave) * 16;

    // cooperative A-load indices: thread t -> (row rr, col cc) and (rr+8, cc)
    const int rr = threadIdx.x >> 5;       // 0..7
    const int cc = threadIdx.x & 31;       // 0..31

    v8f acc = {};
    for (int k0 = 0; k0 < K; k0 += KCHUNK) {
        As[rr][cc]     = A[(size_t)(row0 + rr) * K + k0 + cc];
        As[rr + 8][cc] = A[(size_t)(row0 + rr + 8) * K + k0 + cc];
        __syncthreads();

#pragma unroll
        for (int kk = 0; kk < KCHUNK; kk += 4) {
            const int ka = kk + half * 2;
            v2f a, b;
            a.x = As[l16][ka];
            a.y = As[l16][ka + 1];
            const float* __restrict__ Bp =
                B + (size_t)(k0 + ka) * Ncols + col0 + l16;
            b.x = Bp[0];
            b.y = Bp[Ncols];
            acc = __builtin_amdgcn_wmma_f32_16x16x4_f32(
                /*neg_a=*/false, a, /*neg_b=*/false, b,
                /*c_mod=*/(short)0, acc, /*reuse_a=*/false, /*reuse_b=*/false);
        }
        __syncthreads();
    }

    float* __restrict__ Cp = C + (size_t)(row0 + half * 8) * Ncols + col0 + l16;
#pragma unroll
    for (int j = 0; j < 8; ++j)
        Cp[(size_t)j * Ncols] = acc[j];
}

// ---------------------------------------------------------------------------
// Edge scatter: agg[dst] += h[src] * dinv[src]*dinv[dst]
// 128 threads per edge, float4 each -> coalesced 2KB row gather + atomics.
// ---------------------------------------------------------------------------
__global__ __launch_bounds__(256) void k_scatter(const float* __restrict__ h,
                                                 const int* __restrict__ ei,
                                                 const float* __restrict__ dinv,
                                                 float* __restrict__ agg) {
    const long long t = (long long)blockIdx.x * blockDim.x + threadIdx.x;
    const int e  = (int)(t >> 7);          // HDIM/4 = 128 float4 slots per edge
    const int j  = ((int)t & 127) * 4;
    const int s  = ei[e];
    const int d  = ei[NEDGES + e];
    const float norm = dinv[s] * dinv[d];
    const float4 hv = *(const float4*)(h + (size_t)s * HDIM + j);
    float* ap = agg + (size_t)d * HDIM + j;
    unsafeAtomicAdd(ap + 0, hv.x * norm);
    unsafeAtomicAdd(ap + 1, hv.y * norm);
    unsafeAtomicAdd(ap + 2, hv.z * norm);
    unsafeAtomicAdd(ap + 3, hv.w * norm);
}

// ---------------------------------------------------------------------------
// Fused: v = agg[i] + h[i]*dinv[i]^2 (self loop) + bias; LayerNorm; ReLU
// one block (128 threads, 4 waves) per node, float4 per thread.
// ---------------------------------------------------------------------------
__global__ __launch_bounds__(128) void k_selfloop_ln_relu(const float* __restrict__ agg,
                                                          const float* __restrict__ h,
                                                          const float* __restrict__ dinv,
                                                          const float* __restrict__ bias,
                                                          const float* __restrict__ gamma,
                                                          const float* __restrict__ beta,
                                                          float* __restrict__ out) {
    const int i   = blockIdx.x;
    const int tid = threadIdx.x;
    const int j   = tid * 4;

    const float di   = dinv[i];
    const float self = di * di;

    const float4 av = *(const float4*)(agg  + (size_t)i * HDIM + j);
    const float4 hv = *(const float4*)(h    + (size_t)i * HDIM + j);
    const float4 bv = *(const float4*)(bias + j);

    float4 v;
    v.x = av.x + hv.x * self + bv.x;
    v.y = av.y + hv.y * self + bv.y;
    v.z = av.z + hv.z * self + bv.z;
    v.w = av.w + hv.w * self + bv.w;

    float s  = v.x + v.y + v.z + v.w;
    float ss = v.x * v.x + v.y * v.y + v.z * v.z + v.w * v.w;
#pragma unroll
    for (int o = 16; o > 0; o >>= 1) {
        s  += __shfl_down(s, o);
        ss += __shfl_down(ss, o);
    }
    __shared__ float wsum[4], wsq[4], stats[2];
    if ((tid & 31) == 0) { wsum[tid >> 5] = s; wsq[tid >> 5] = ss; }
    __syncthreads();
    if (tid == 0) {
        float ts  = wsum[0] + wsum[1] + wsum[2] + wsum[3];
        float tss = wsq[0] + wsq[1] + wsq[2] + wsq[3];
        float mu  = ts * (1.0f / HDIM);
        float var = tss * (1.0f / HDIM) - mu * mu;
        stats[0] = mu;
        stats[1] = rsqrtf(var + LNEPS);
    }
    __syncthreads();
    const float mu = stats[0], rs = stats[1];

    const float4 gv  = *(const float4*)(gamma + j);
    const float4 btv = *(const float4*)(beta + j);
    float4 r;
    r.x = fmaxf((v.x - mu) * rs * gv.x + btv.x, 0.0f);
    r.y = fmaxf((v.y - mu) * rs * gv.y + btv.y, 0.0f);
    r.z = fmaxf((v.z - mu) * rs * gv.z + btv.z, 0.0f);
    r.w = fmaxf((v.w - mu) * rs * gv.w + btv.w, 0.0f);
    *(float4*)(out + (size_t)i * HDIM + j) = r;
}

// ---------------------------------------------------------------------------
// Mean pool, stage 1: row-chunked partial sums, fully coalesced float4 reads.
// 160 blocks x 125 rows; partials accumulated into pre-zeroed g via
// global_atomic_add_f32 with 1/N folded in.
// ---------------------------------------------------------------------------
__global__ __launch_bounds__(128) void k_colmean_partial(const float* __restrict__ x,
                                                         float* __restrict__ g) {
    const int r0 = blockIdx.x * MPROWS;
    const int j  = threadIdx.x * 4;
    float4 s = {0.0f, 0.0f, 0.0f, 0.0f};
    for (int i = 0; i < MPROWS; ++i) {
        const float4 v = *(const float4*)(x + (size_t)(r0 + i) * HDIM + j);
        s.x += v.x; s.y += v.y; s.z += v.z; s.w += v.w;
    }
    const float sc = 1.0f / (float)NNODES;
    float* gp = g + j;
    unsafeAtomicAdd(gp + 0, s.x * sc);
    unsafeAtomicAdd(gp + 1, s.y * sc);
    unsafeAtomicAdd(gp + 2, s.z * sc);
    unsafeAtomicAdd(gp + 3, s.w * sc);
}

// ---------------------------------------------------------------------------
// out[o] = b_out[o] + sum_j g[j] * W_out[j, o]
// ---------------------------------------------------------------------------
__global__ void k_out(const float* __restrict__ g, const float* __restrict__ Wout,
                      const float* __restrict__ bout, float* __restrict__ out) {
    const int o = threadIdx.x;
    if (o >= NOUT) return;
    float acc = bout[o];
    for (int j = 0; j < HDIM; ++j)
        acc += g[j] * Wout[j * NOUT + o];
    out[o] = acc;
}

// ---------------------------------------------------------------------------
// Host-side orchestration
// ---------------------------------------------------------------------------
extern "C" void kernel_launch(void* const* d_in, const int* in_sizes, int n_in,
                              void* d_out, int out_size, void* d_ws, size_t ws_size,
                              hipStream_t stream) {
    const float* x    = (const float*)d_in[0];
    const int*   ei   = (const int*)d_in[1];
    const float* W[4] = {(const float*)d_in[2], (const float*)d_in[4],
                         (const float*)d_in[6], (const float*)d_in[8]};
    const float* b[4] = {(const float*)d_in[3], (const float*)d_in[5],
                         (const float*)d_in[7], (const float*)d_in[9]};
    const float* gamma = (const float*)d_in[10];
    const float* beta  = (const float*)d_in[11];
    const float* Wout  = (const float*)d_in[12];
    const float* bout  = (const float*)d_in[13];
    float* out = (float*)d_out;

    const size_t NH = (size_t)NNODES * HDIM;            // 10.24M floats
    char* ws = (char*)d_ws;
    float* h_buf   = (float*)(ws);
    float* agg_buf = (float*)(ws + NH * sizeof(float));
    float* x_buf   = (float*)(ws + 2 * NH * sizeof(float));
    float* dinv    = (float*)(ws + 3 * NH * sizeof(float));
    float* g_buf   = dinv + NNODES;

    // degrees (with self loops) -> dinv, in place
    k_init_deg<<<(NNODES + 255) / 256, 256, 0, stream>>>(dinv, NNODES);
    k_deg_edges<<<(NEDGES + 255) / 256, 256, 0, stream>>>(ei, dinv, NEDGES);
    k_dinv<<<(NNODES + 255) / 256, 256, 0, stream>>>(dinv, NNODES);

    const dim3 gemm_block(256);
    const dim3 gemm_grid(NNODES / 16, HDIM / (16 * 8));       // 1250 x 4
    const int  scatter_blocks = (NEDGES * (HDIM / 4)) / 256;  // 160000 exact

    const float* x_cur = x;
    int K = DIN;
    for (int l = 0; l < 4; ++l) {
        // h = x_cur @ W[l]
        k_gemm_wmma_f32<<<gemm_grid, gemm_block, 0, stream>>>(x_cur, W[l], h_buf, K, HDIM);
        // agg = 0
        hipMemsetAsync(agg_buf, 0, NH * sizeof(float), stream);
        // agg[dst] += h[src] * dinv[src]*dinv[dst]
        k_scatter<<<scatter_blocks, 256, 0, stream>>>(h_buf, ei, dinv, agg_buf);
        // x_buf = relu(LN(agg + self + bias))
        k_selfloop_ln_relu<<<NNODES, 128, 0, stream>>>(agg_buf, h_buf, dinv,
                                                       b[l], gamma, beta, x_buf);
        x_cur = x_buf;
        K = HDIM;
    }

    hipMemsetAsync(g_buf, 0, HDIM * sizeof(float), stream);
    k_colmean_partial<<<NNODES / MPROWS, 128, 0, stream>>>(x_buf, g_buf);
    k_out<<<1, 32, 0, stream>>>(g_buf, Wout, bout, out);
}